// GAT_6828998001549
// MI455X (gfx1250) — compile-verified
//
#include <hip/hip_runtime.h>
#include <math.h>

// ---------------------------------------------------------------------------
// GAT (2-layer, PyG-style) for MI455X / gfx1250.
// Dense GEMMs: V_WMMA_F32_16X16X4_F32 with B staged in LDS (one cooperative
// fill per block, reused by 8 waves x 1 row-tile), A software-pipelined from
// global. Weights pre-transposed + zero-padded so all accesses are
// unconditional. Edge softmax: 3-pass segment reduction, L2-resident atomics.
// ---------------------------------------------------------------------------

typedef __attribute__((ext_vector_type(2))) float v2f;
typedef __attribute__((ext_vector_type(8))) float v8f;

#define NNODES   50000
#define INCH     128
#define HEADS1   8
#define HID      32
#define F1       256     // HEADS1*HID
#define OUTCH    40
#define NCPAD2   48      // OUTCH padded to tile multiple
#define NEGSLOPE 0.2f
#define WPB      8       // waves per block (256 threads)

// ---------------------------------------------------------------------------
// generic fill
// ---------------------------------------------------------------------------
__global__ void k_fill(float* __restrict__ p, int n, float v) {
    int i = blockIdx.x * blockDim.x + threadIdx.x;
    if (i < n) p[i] = v;
}

// ---------------------------------------------------------------------------
// Wt[c*K + k] = (c < Ncols) ? W[k*Ncols + c] : 0   (transpose + column pad)
// ---------------------------------------------------------------------------
__global__ void k_transpose_pad(const float* __restrict__ W, float* __restrict__ Wt,
                                int K, int Ncols, int NcPad) {
    int i = blockIdx.x * blockDim.x + threadIdx.x;   // over NcPad*K
    if (i >= NcPad * K) return;
    const int c = i / K;
    const int k = i - c * K;
    Wt[i] = (c < Ncols) ? W[(size_t)k * Ncols + c] : 0.0f;
}

// ---------------------------------------------------------------------------
// WMMA fp32 GEMM: D[M x Ncols] = A[M x KK] @ B[KK x Ncols]
// Bt: transposed+padded weights [NcPad x KK]. Block = 8 waves; blockIdx.y
// selects a 16*NT-column strip whose weights are staged once in LDS; wave w
// computes row-tile blockIdx.x*8+w against that strip. Inner loop:
// 1 pipelined global b64 (A) + NT ds b64 (B) + NT WMMAs. Row stride KK+2
// keeps LDS bank-conflict-free. M must be a multiple of 16.
// ---------------------------------------------------------------------------
template <int NT, int KK>
__global__ void k_gemm_wmma_lds(const float* __restrict__ A,
                                const float* __restrict__ Bt,
                                float* __restrict__ D,
                                int M, int Ncols) {
    constexpr int COLS = 16 * NT;      // strip width
    constexpr int LDSROW = KK + 2;     // padded LDS row stride (banks +2)
    __shared__ float Bl[COLS * LDSROW];

    // ---- cooperative LDS fill: strip columns of Bt ----
    const int cbase = blockIdx.y * COLS;
    for (int idx = threadIdx.x; idx < COLS * KK; idx += blockDim.x) {
        const int c = idx / KK;
        const int k = idx - c * KK;
        Bl[c * LDSROW + k] = Bt[(size_t)(cbase + c) * KK + k];
    }
    __syncthreads();

    const int wv  = threadIdx.x >> 5;
    const int lane = threadIdx.x & 31;
    const int tm = blockIdx.x * WPB + wv;
    if (tm >= (M >> 4)) return;        // wave-uniform, after the barrier

    const int half = lane >> 4;        // 0: K {0,1}, 1: K {2,3}
    const int l16  = lane & 15;
    const int koff = half << 1;

    const int arow = tm * 16 + l16;
    const float* __restrict__ Ap = A + (size_t)arow * KK + koff;

    const float* Blp[NT];
    v8f acc[NT];
    #pragma unroll
    for (int t = 0; t < NT; ++t) {
        Blp[t] = &Bl[(t * 16 + l16) * LDSROW + koff];
        acc[t] = (v8f){};
    }

    // ---- K loop: A pipelined from global, B from LDS ----
    v2f a_cur = *(const v2f*)(Ap);
    #pragma unroll 4
    for (int k = 0; k < KK - 4; k += 4) {
        const v2f a_nxt = *(const v2f*)(Ap + k + 4);
        #pragma unroll
        for (int t = 0; t < NT; ++t) {
            const v2f b = *(const v2f*)(Blp[t] + k);         // ds_load_b64
            acc[t] = __builtin_amdgcn_wmma_f32_16x16x4_f32(
                false, a_cur, false, b, (short)0, acc[t], false, false);
        }
        a_cur = a_nxt;
    }
    #pragma unroll
    for (int t = 0; t < NT; ++t) {
        const v2f b = *(const v2f*)(Blp[t] + KK - 4);
        acc[t] = __builtin_amdgcn_wmma_f32_16x16x4_f32(
            false, a_cur, false, b, (short)0, acc[t], false, false);
    }

    // ---- store 16x16 tiles (guard padded columns) ----
    #pragma unroll
    for (int t = 0; t < NT; ++t) {
        const int bcol = cbase + t * 16 + l16;
        if (bcol < Ncols) {
            #pragma unroll
            for (int r = 0; r < 8; ++r) {
                const int orow = tm * 16 + (half << 3) + r;
                D[(size_t)orow * Ncols + bcol] = acc[t][r];
            }
        }
    }
}

// ---------------------------------------------------------------------------
// Per-(node,head) attention logits: as[n,h] = <H[n,h,:], att_src[h,:]>, etc.
// ---------------------------------------------------------------------------
__global__ void k_attn_dots(const float* __restrict__ H,
                            const float* __restrict__ att_src,
                            const float* __restrict__ att_dst,
                            float* __restrict__ as_, float* __restrict__ ad_,
                            int N, int Hh, int C) {
    int i = blockIdx.x * blockDim.x + threadIdx.x;
    if (i >= N * Hh) return;
    const int h = i % Hh;
    const float* __restrict__ hp = H + (size_t)i * C;
    const float* __restrict__ sp = att_src + (size_t)h * C;
    const float* __restrict__ dp = att_dst + (size_t)h * C;
    float vs = 0.0f, vd = 0.0f;
    for (int c = 0; c < C; ++c) {
        const float x = hp[c];
        vs = fmaf(x, sp[c], vs);
        vd = fmaf(x, dp[c], vd);
    }
    as_[i] = vs;
    ad_[i] = vd;
}

// ---------------------------------------------------------------------------
// Edge passes. Edge e < E uses edge_index; e >= E is the self-loop (e-E,e-E).
// ---------------------------------------------------------------------------
__device__ __forceinline__ float leaky(float v) {
    return v > 0.0f ? v : NEGSLOPE * v;
}

__global__ void k_edge_max(const int* __restrict__ srcI, const int* __restrict__ dstI,
                           int E, int N, int Hh,
                           const float* __restrict__ as_, const float* __restrict__ ad_,
                           float* __restrict__ emax) {
    int i = blockIdx.x * blockDim.x + threadIdx.x;
    const int tot = (E + N) * Hh;
    if (i >= tot) return;
    const int e = i / Hh, h = i - e * Hh;
    const int s = (e < E) ? srcI[e] : (e - E);
    const int d = (e < E) ? dstI[e] : (e - E);
    const float v = leaky(as_[s * Hh + h] + ad_[d * Hh + h]);
    atomicMax(&emax[d * Hh + h], v);           // global_atomic_max_num_f32
}

__global__ void k_edge_expsum(const int* __restrict__ srcI, const int* __restrict__ dstI,
                              int E, int N, int Hh,
                              const float* __restrict__ as_, const float* __restrict__ ad_,
                              const float* __restrict__ emax,
                              float* __restrict__ denom) {
    int i = blockIdx.x * blockDim.x + threadIdx.x;
    const int tot = (E + N) * Hh;
    if (i >= tot) return;
    const int e = i / Hh, h = i - e * Hh;
    const int s = (e < E) ? srcI[e] : (e - E);
    const int d = (e < E) ? dstI[e] : (e - E);
    const float v = leaky(as_[s * Hh + h] + ad_[d * Hh + h]);
    const float ee = __expf(v - emax[d * Hh + h]);
    atomicAdd(&denom[d * Hh + h], ee);
}

__global__ void k_edge_scatter(const int* __restrict__ srcI, const int* __restrict__ dstI,
                               int E, int N, int Hh, int C,
                               const float* __restrict__ as_, const float* __restrict__ ad_,
                               const float* __restrict__ emax, const float* __restrict__ denom,
                               const float* __restrict__ Hfeat,
                               float* __restrict__ outAcc) {
    int i = blockIdx.x * blockDim.x + threadIdx.x;
    const int tot = (E + N) * Hh;
    if (i >= tot) return;
    const int e = i / Hh, h = i - e * Hh;
    const int s = (e < E) ? srcI[e] : (e - E);
    const int d = (e < E) ? dstI[e] : (e - E);
    const float v  = leaky(as_[s * Hh + h] + ad_[d * Hh + h]);
    const float ee = __expf(v - emax[d * Hh + h]);
    const float alpha = ee / (denom[d * Hh + h] + 1e-16f);
    const float* __restrict__ hp = Hfeat + ((size_t)s * Hh + h) * C;
    float* __restrict__ op = outAcc + ((size_t)d * Hh + h) * C;
    for (int c = 0; c < C; ++c)
        atomicAdd(&op[c], hp[c] * alpha);      // L2-resident global_atomic_add_f32
}

// ---------------------------------------------------------------------------
// Epilogues
// ---------------------------------------------------------------------------
__global__ void k_bias_elu(float* __restrict__ p, const float* __restrict__ bias,
                           int N, int F) {
    int i = blockIdx.x * blockDim.x + threadIdx.x;
    if (i >= N * F) return;
    const float v = p[i] + bias[i % F];
    p[i] = v > 0.0f ? v : (__expf(v) - 1.0f);
}

__global__ void k_bias_out(const float* __restrict__ acc, const float* __restrict__ bias,
                           float* __restrict__ out, int N, int F) {
    int i = blockIdx.x * blockDim.x + threadIdx.x;
    if (i >= N * F) return;
    out[i] = acc[i] + bias[i % F];
}

// ---------------------------------------------------------------------------
// Host-side orchestration
// ---------------------------------------------------------------------------
static inline int cdiv(int a, int b) { return (a + b - 1) / b; }

extern "C" void kernel_launch(void* const* d_in, const int* in_sizes, int n_in,
                              void* d_out, int out_size, void* d_ws, size_t ws_size,
                              hipStream_t stream) {
    const float* x        = (const float*)d_in[0];   // [N,128]
    const int*   ei       = (const int*)  d_in[1];   // [2,E]
    const float* W1       = (const float*)d_in[2];   // [128,256]
    const float* att_src1 = (const float*)d_in[3];   // [8,32]
    const float* att_dst1 = (const float*)d_in[4];   // [8,32]
    const float* b1       = (const float*)d_in[5];   // [256]
    const float* W2       = (const float*)d_in[6];   // [256,40]
    const float* att_src2 = (const float*)d_in[7];   // [1,40]
    const float* att_dst2 = (const float*)d_in[8];   // [1,40]
    const float* b2       = (const float*)d_in[9];   // [40]
    float* out = (float*)d_out;

    const int N = in_sizes[0] / INCH;                // 50000
    const int E = in_sizes[1] / 2;                   // 800000
    const int* srcI = ei;
    const int* dstI = ei + E;

    // ---- workspace layout (fp32) ----
    float* ws = (float*)d_ws;
    float* h1    = ws;               ws += (size_t)N * F1;     // X@W1
    float* x2    = ws;               ws += (size_t)N * F1;     // L1 out -> elu -> L2 in
    float* as1   = ws;               ws += (size_t)N * HEADS1;
    float* ad1   = ws;               ws += (size_t)N * HEADS1;
    float* emax1 = ws;               ws += (size_t)N * HEADS1;
    float* den1  = ws;               ws += (size_t)N * HEADS1;
    float* h2    = ws;               ws += (size_t)N * OUTCH;  // X2@W2
    float* as2   = ws;               ws += (size_t)N;
    float* ad2   = ws;               ws += (size_t)N;
    float* emax2 = ws;               ws += (size_t)N;
    float* den2  = ws;               ws += (size_t)N;
    float* out2  = ws;               ws += (size_t)N * OUTCH;  // L2 accumulator
    float* W1t   = ws;               ws += (size_t)F1 * INCH;  // [256 x 128] transposed
    float* W2t   = ws;               ws += (size_t)NCPAD2 * F1;// [48 x 256] transposed+pad

    const int TB = 256;
    const float NEG_INF = -INFINITY;

    // ---- init accumulators (fresh every call: deterministic) ----
    k_fill<<<cdiv(N * HEADS1, TB), TB, 0, stream>>>(emax1, N * HEADS1, NEG_INF);
    k_fill<<<cdiv(N * HEADS1, TB), TB, 0, stream>>>(den1,  N * HEADS1, 0.0f);
    k_fill<<<cdiv(N * F1,     TB), TB, 0, stream>>>(x2,    N * F1,     0.0f);
    k_fill<<<cdiv(N,          TB), TB, 0, stream>>>(emax2, N,          NEG_INF);
    k_fill<<<cdiv(N,          TB), TB, 0, stream>>>(den2,  N,          0.0f);
    k_fill<<<cdiv(N * OUTCH,  TB), TB, 0, stream>>>(out2,  N * OUTCH,  0.0f);

    // ---- weight transpose + pad (tiny, L2-resident) ----
    k_transpose_pad<<<cdiv(F1 * INCH,    TB), TB, 0, stream>>>(W1, W1t, INCH, F1, F1);
    k_transpose_pad<<<cdiv(NCPAD2 * F1,  TB), TB, 0, stream>>>(W2, W2t, F1, OUTCH, NCPAD2);

    // ---- layer 1: h1 = x @ W1 : 4 strips of 64 cols, 8 row-tiles/block ----
    {
        dim3 grid(cdiv(N / 16, WPB), F1 / 64);           // (391, 4)
        k_gemm_wmma_lds<4, INCH><<<grid, TB, 0, stream>>>(x, W1t, h1, N, F1);
    }

    // attention logits
    k_attn_dots<<<cdiv(N * HEADS1, TB), TB, 0, stream>>>(
        h1, att_src1, att_dst1, as1, ad1, N, HEADS1, HID);

    // segment softmax + scatter (layer 1)
    {
        const int tot = (E + N) * HEADS1;
        k_edge_max    <<<cdiv(tot, TB), TB, 0, stream>>>(srcI, dstI, E, N, HEADS1, as1, ad1, emax1);
        k_edge_expsum <<<cdiv(tot, TB), TB, 0, stream>>>(srcI, dstI, E, N, HEADS1, as1, ad1, emax1, den1);
        k_edge_scatter<<<cdiv(tot, TB), TB, 0, stream>>>(srcI, dstI, E, N, HEADS1, HID,
                                                         as1, ad1, emax1, den1, h1, x2);
    }

    // x2 = elu(x2 + b1)
    k_bias_elu<<<cdiv(N * F1, TB), TB, 0, stream>>>(x2, b1, N, F1);

    // ---- layer 2: h2 = x2 @ W2 : 1 strip of 48 cols, 8 row-tiles/block ----
    {
        dim3 grid(cdiv(N / 16, WPB), 1);                 // (391, 1)
        k_gemm_wmma_lds<3, F1><<<grid, TB, 0, stream>>>(x2, W2t, h2, N, OUTCH);
    }

    k_attn_dots<<<cdiv(N, TB), TB, 0, stream>>>(
        h2, att_src2, att_dst2, as2, ad2, N, 1, OUTCH);

    {
        const int tot = (E + N) * 1;
        k_edge_max    <<<cdiv(tot, TB), TB, 0, stream>>>(srcI, dstI, E, N, 1, as2, ad2, emax2);
        k_edge_expsum <<<cdiv(tot, TB), TB, 0, stream>>>(srcI, dstI, E, N, 1, as2, ad2, emax2, den2);
        k_edge_scatter<<<cdiv(tot, TB), TB, 0, stream>>>(srcI, dstI, E, N, 1, OUTCH,
                                                         as2, ad2, emax2, den2, h2, out2);
    }

    // heads=1, concat=False -> mean over 1 head == identity; add bias.
    k_bias_out<<<cdiv(N * OUTCH, TB), TB, 0, stream>>>(out2, b2, out, N, OUTCH);
}